// MemoryBank_47571057770864
// MI455X (gfx1250) — compile-verified
//
#include <hip/hip_runtime.h>

typedef __attribute__((ext_vector_type(2))) float v2f;
typedef __attribute__((ext_vector_type(8))) float v8f;

#define NCHUNK 256
#define PSTRIDE 1088  // 32*32 cross partial + 64 sq-partial slots per chunk

// ---------------------------------------------------------------------------
// Kernel A: split-K weighted cross-correlation GEMM via V_WMMA_F32_16X16X4_F32
//   score_partial[b,m] += sum_l (-2/K_l) * f_l[b,k]*mem_l[m,k]  (k in chunk)
//   sq_partial(lane)   += sum_l ( 1/K_l) * mem_l[m,k]^2
// One wave (32 threads) per chunk; 4 accumulator tiles cover the 32x32 output.
// ---------------------------------------------------------------------------
__global__ __launch_bounds__(32) void mb_cross_kernel(
    const float* __restrict__ f1, const float* __restrict__ m1,
    const float* __restrict__ f2, const float* __restrict__ m2,
    const float* __restrict__ f3, const float* __restrict__ m3,
    float* __restrict__ partials)
{
  const int chunk = blockIdx.x;
  const int lane  = threadIdx.x;          // 0..31
  const int r     = lane & 15;            // row/col within 16-tile
  const int khalf = (lane >> 4) << 1;     // lanes 0-15 hold k0,k0+1 ; 16-31 hold k0+2,k0+3
  const bool mt1_valid = (16 + r) < 30;   // mem rows 30,31 are padding

  v8f acc00 = {}; v8f acc01 = {}; v8f acc10 = {}; v8f acc11 = {};
  float sq0 = 0.f, sq1 = 0.f;

  const float* Fs[3] = {f1, f2, f3};
  const float* Ms[3] = {m1, m2, m3};
  const int    Ks[3] = {64 * 64 * 64, 128 * 32 * 32, 256 * 16 * 16};

  for (int l = 0; l < 3; ++l) {
    const int K = Ks[l];
    const float wl = -2.0f / (float)K;   // layer weight folded into A fragment
    const float sl =  1.0f / (float)K;   // layer weight for m2 accumulation
    const int span = K / NCHUNK;
    const int kbeg = chunk * span;
    const float* __restrict__ F = Fs[l];
    const float* __restrict__ M = Ms[l];

    for (int k0 = kbeg; k0 < kbeg + span; k0 += 4) {
      const int kk = k0 + khalf;
      v2f a0 = *(const v2f*)(F + (size_t)r * K + kk);         // f rows 0-15
      v2f a1 = *(const v2f*)(F + (size_t)(r + 16) * K + kk);  // f rows 16-31
      v2f b0 = *(const v2f*)(M + (size_t)r * K + kk);         // mem rows 0-15
      v2f b1 = {};                                            // mem rows 16-29 (+pad)
      if (mt1_valid) b1 = *(const v2f*)(M + (size_t)(r + 16) * K + kk);

      sq0 += sl * (b0.x * b0.x + b0.y * b0.y);
      sq1 += sl * (b1.x * b1.x + b1.y * b1.y);

      a0.x *= wl; a0.y *= wl;
      a1.x *= wl; a1.y *= wl;

      acc00 = __builtin_amdgcn_wmma_f32_16x16x4_f32(false, a0, false, b0, (short)0, acc00, false, false);
      acc01 = __builtin_amdgcn_wmma_f32_16x16x4_f32(false, a0, false, b1, (short)0, acc01, false, false);
      acc10 = __builtin_amdgcn_wmma_f32_16x16x4_f32(false, a1, false, b0, (short)0, acc10, false, false);
      acc11 = __builtin_amdgcn_wmma_f32_16x16x4_f32(false, a1, false, b1, (short)0, acc11, false, false);
    }
  }

  // C/D layout: VGPR j, lanes 0-15 -> (M=j, N=lane); lanes 16-31 -> (M=j+8, N=lane-16)
  float* __restrict__ P = partials + (size_t)chunk * PSTRIDE;
  const int n = lane & 15;
  const int rowOff = (lane >> 4) * 8;
#pragma unroll
  for (int j = 0; j < 8; ++j) {
    P[(rowOff + j) * 32 + n]             = acc00[j];
    P[(rowOff + j) * 32 + (n + 16)]      = acc01[j];
    P[(rowOff + j + 16) * 32 + n]        = acc10[j];
    P[(rowOff + j + 16) * 32 + (n + 16)] = acc11[j];
  }
  P[1024 + lane]      = sq0;  // per-lane sq partial, m-tile 0
  P[1024 + 32 + lane] = sq1;  // per-lane sq partial, m-tile 1
}

// ---------------------------------------------------------------------------
// Kernel B: fixed-order reduction of partials, add m2[m], argmin over m<30.
// ---------------------------------------------------------------------------
__global__ __launch_bounds__(1024) void mb_reduce_kernel(
    const float* __restrict__ partials, int* __restrict__ idx)
{
  __shared__ float score[1024];
  __shared__ float raw[64];
  __shared__ float m2s[32];
  const int t = threadIdx.x;

  float s = 0.f;
  for (int c = 0; c < NCHUNK; ++c) s += partials[(size_t)c * PSTRIDE + t];
  score[t] = s;

  if (t < 64) {
    float q = 0.f;
    for (int c = 0; c < NCHUNK; ++c) q += partials[(size_t)c * PSTRIDE + 1024 + t];
    raw[t] = q;
  }
  __syncthreads();

  if (t < 32) {
    // m < 16: lanes m and m+16 of tile0 ; m >= 16: lanes (m-16) and (m-16)+16 of tile1
    m2s[t] = (t < 16) ? (raw[t] + raw[t + 16]) : (raw[t + 16] + raw[t + 32]);
  }
  __syncthreads();

  if (t < 32) {
    float best = score[t * 32 + 0] + m2s[0];
    int bi = 0;
    for (int m = 1; m < 30; ++m) {
      float v = score[t * 32 + m] + m2s[m];
      if (v < best) { best = v; bi = m; }  // strict < keeps first-min like argmin
    }
    idx[t] = bi;
  }
}

// ---------------------------------------------------------------------------
// Kernel C: out[b] = concat(f[b], (mem[idx[b]] - f[b])^2) along channels.
// Pure bandwidth; float4 vectorized (HW is a multiple of 4 for all layers).
// ---------------------------------------------------------------------------
__global__ __launch_bounds__(256) void mb_out_kernel(
    const float* __restrict__ f, const float* __restrict__ mem,
    const int* __restrict__ idx, float* __restrict__ out,
    int C, int HW4, long total4)
{
  const float4* __restrict__ F4 = (const float4*)f;
  const float4* __restrict__ M4 = (const float4*)mem;
  float4* __restrict__ O4 = (float4*)out;
  const long stride = (long)gridDim.x * blockDim.x;

  for (long i = (long)blockIdx.x * blockDim.x + threadIdx.x; i < total4; i += stride) {
    long hw = i % HW4;
    long tc = i / HW4;
    int c2 = (int)(tc % (2 * C));
    int b  = (int)(tc / (2 * C));
    if (c2 < C) {
      O4[i] = F4[((long)b * C + c2) * HW4 + hw];
    } else {
      int c = c2 - C;
      float4 fv = F4[((long)b * C + c) * HW4 + hw];
      float4 mv = M4[((long)idx[b] * C + c) * HW4 + hw];
      float4 d;
      d.x = mv.x - fv.x; d.y = mv.y - fv.y; d.z = mv.z - fv.z; d.w = mv.w - fv.w;
      d.x *= d.x; d.y *= d.y; d.z *= d.z; d.w *= d.w;
      O4[i] = d;
    }
  }
}

// ---------------------------------------------------------------------------
extern "C" void kernel_launch(void* const* d_in, const int* in_sizes, int n_in,
                              void* d_out, int out_size, void* d_ws, size_t ws_size,
                              hipStream_t stream) {
  (void)in_sizes; (void)n_in; (void)out_size; (void)ws_size;
  // setup_inputs() dict order: f1, mem1, f2, mem2, f3, mem3
  const float* f1 = (const float*)d_in[0];
  const float* m1 = (const float*)d_in[1];
  const float* f2 = (const float*)d_in[2];
  const float* m2 = (const float*)d_in[3];
  const float* f3 = (const float*)d_in[4];
  const float* m3 = (const float*)d_in[5];
  float* out = (float*)d_out;

  float* partials = (float*)d_ws;
  int* idx = (int*)((char*)d_ws + (size_t)NCHUNK * PSTRIDE * sizeof(float));

  mb_cross_kernel<<<NCHUNK, 32, 0, stream>>>(f1, m1, f2, m2, f3, m3, partials);
  mb_reduce_kernel<<<1, 1024, 0, stream>>>(partials, idx);

  // layer 1: [32,128,64,64]  layer 2: [32,256,32,32]  layer 3: [32,512,16,16]
  mb_out_kernel<<<2048, 256, 0, stream>>>(f1, m1, idx, out,            64,  1024, 4194304L);
  mb_out_kernel<<<1024, 256, 0, stream>>>(f2, m2, idx, out + 16777216, 128, 256,  2097152L);
  mb_out_kernel<<<512,  256, 0, stream>>>(f3, m3, idx, out + 25165824, 256, 64,   1048576L);
}